// SparseGraphWaveletLayer_17952963297709
// MI455X (gfx1250) — compile-verified
//
#include <hip/hip_runtime.h>
#include <hip/hip_bf16.h>

typedef float v2f __attribute__((ext_vector_type(2)));
typedef float v8f __attribute__((ext_vector_type(8)));

#define N_NODES 50000
#define IN_CH   256
#define OUT_CH  128
#define NNZ_CNT 800000

// ---------------------------------------------------------------------------
// Deterministic native f32 atomic add (no-return form, STOREcnt-tracked).
// Avoids any chance of hipcc expanding atomicAdd(float*) into a CAS loop.
// ---------------------------------------------------------------------------
__device__ __forceinline__ void atomic_add_f32(float* addr, float val) {
    asm volatile("global_atomic_add_f32 %0, %1, off"
                 :
                 : "v"(addr), "v"(val)
                 : "memory");
}

// ---------------------------------------------------------------------------
// Zero a buffer (float4 granularity, grid-stride)
// ---------------------------------------------------------------------------
__global__ void zero4_kernel(float4* __restrict__ p, long n4) {
    long i = (long)blockIdx.x * blockDim.x + threadIdx.x;
    long stride = (long)gridDim.x * blockDim.x;
    float4 z = make_float4(0.f, 0.f, 0.f, 0.f);
    for (; i < n4; i += stride) p[i] = z;
}

// ---------------------------------------------------------------------------
// Densify the sparse feature matrix: dense[r*256 + c] += v
// ---------------------------------------------------------------------------
__global__ void scatter_features_kernel(const int* __restrict__ rows,
                                        const int* __restrict__ cols,
                                        const float* __restrict__ vals,
                                        float* __restrict__ dense, int nnz) {
    int i = blockIdx.x * blockDim.x + threadIdx.x;
    if (i < nnz) {
        atomic_add_f32(dense + (long)rows[i] * IN_CH + cols[i], vals[i]);
    }
}

// ---------------------------------------------------------------------------
// Dense GEMM via V_WMMA_F32_16X16X4_F32:
//   C[50000x128] = A[50000x256] * B[256x128]
// One wave computes a full 16x128 strip (8 accumulator tiles) so each A
// fragment is loaded once and fed to 8 WMMAs. 3125 strips total.
// blockDim = 128 (4 waves/block); EXEC all-ones (guards are wave-uniform).
// ---------------------------------------------------------------------------
__global__ __launch_bounds__(128)
void gemm_wmma_kernel(const float* __restrict__ A,   // [M x 256]
                      const float* __restrict__ B,   // [256 x 128]
                      float* __restrict__ C,         // [M x 128]
                      int M) {
    const int strip = blockIdx.x * (blockDim.x >> 5) + (threadIdx.x >> 5);
    const int lane  = threadIdx.x & 31;
    if (strip >= M / 16) return;             // wave-uniform

    const int half = lane >> 4;              // 0: lanes 0-15, 1: lanes 16-31
    const int l16  = lane & 15;

    // ISA 32-bit layouts (16x16x4):
    //   A: VGPR0 = A[strip*16+l16][k0 + 2*half], VGPR1 = same row, K+1
    //   B: VGPR0 = B[k0 + 2*half][col], VGPR1 = K+1 row, same col
    const long rowA = (long)(strip * 16 + l16) * IN_CH;

    v8f acc[8] = {};
    #pragma unroll 2
    for (int k0 = 0; k0 < IN_CH; k0 += 4) {
        const int ka = k0 + 2 * half;
        const v2f a = *(const v2f*)(A + rowA + ka);   // (K, K+1), 8B aligned
        #pragma unroll
        for (int t = 0; t < 8; ++t) {
            v2f b;
            b.x = B[(long)ka * OUT_CH + t * 16 + l16];
            b.y = B[(long)(ka + 1) * OUT_CH + t * 16 + l16];
            acc[t] = __builtin_amdgcn_wmma_f32_16x16x4_f32(
                /*neg_a=*/false, a, /*neg_b=*/false, b,
                /*c_mod=*/(short)0, acc[t], /*reuse_a=*/false, /*reuse_b=*/false);
        }
    }

    // C/D layout: VGPR v -> row strip*16 + v + 8*half, col t*16 + l16
    #pragma unroll
    for (int t = 0; t < 8; ++t) {
        #pragma unroll
        for (int v = 0; v < 8; ++v) {
            const int row = strip * 16 + v + 8 * half;
            C[(long)row * OUT_CH + t * 16 + l16] = acc[t][v];
        }
    }
}

// ---------------------------------------------------------------------------
// COO SpMM scatter: dst[r, :] += (v * theta?[c]) * src[c, :]
// One wave per nnz; each lane handles 4 consecutive features (float4 gather,
// coalesced 512B per wave), then 4 native f32 atomics.
// ---------------------------------------------------------------------------
__global__ __launch_bounds__(256)
void spmm_scatter_kernel(const int* __restrict__ rows,
                         const int* __restrict__ cols,
                         const float* __restrict__ vals,
                         const float* __restrict__ theta,  // may be nullptr
                         const float* __restrict__ src,    // [N x 128]
                         float* __restrict__ dst,          // [N x 128]
                         int nnz) {
    const int e    = blockIdx.x * (blockDim.x >> 5) + (threadIdx.x >> 5);
    const int lane = threadIdx.x & 31;
    if (e >= nnz) return;                     // wave-uniform

    const int r = rows[e];
    const int c = cols[e];
    float v = vals[e];
    if (theta) v *= theta[c];                 // wave-uniform branch

    const float4 s = *(const float4*)(src + (long)c * OUT_CH + lane * 4);
    float* d = dst + (long)r * OUT_CH + lane * 4;
    atomic_add_f32(d + 0, v * s.x);
    atomic_add_f32(d + 1, v * s.y);
    atomic_add_f32(d + 2, v * s.z);
    atomic_add_f32(d + 3, v * s.w);
}

// ---------------------------------------------------------------------------
// In-place ReLU (float4 granularity)
// ---------------------------------------------------------------------------
__global__ void relu4_kernel(float4* __restrict__ p, long n4) {
    long i = (long)blockIdx.x * blockDim.x + threadIdx.x;
    long stride = (long)gridDim.x * blockDim.x;
    for (; i < n4; i += stride) {
        float4 x = p[i];
        x.x = fmaxf(x.x, 0.f); x.y = fmaxf(x.y, 0.f);
        x.z = fmaxf(x.z, 0.f); x.w = fmaxf(x.w, 0.f);
        p[i] = x;
    }
}

// ---------------------------------------------------------------------------
// Launch
// ---------------------------------------------------------------------------
extern "C" void kernel_launch(void* const* d_in, const int* in_sizes, int n_in,
                              void* d_out, int out_size, void* d_ws, size_t ws_size,
                              hipStream_t stream) {
    const int*   phi_idx    = (const int*)  d_in[0];   // [2*NNZ] rows then cols
    const float* phi_vals   = (const float*)d_in[1];
    const int*   phinv_idx  = (const int*)  d_in[2];
    const float* phinv_vals = (const float*)d_in[3];
    const int*   feat_idx   = (const int*)  d_in[4];
    const float* feat_vals  = (const float*)d_in[5];
    const float* W          = (const float*)d_in[6];   // [256 x 128]
    const float* theta      = (const float*)d_in[7];   // [N]
    (void)in_sizes; (void)n_in; (void)out_size; (void)ws_size;

    const long denseElems = (long)N_NODES * IN_CH;     // 12.8M floats
    const long featElems  = (long)N_NODES * OUT_CH;    // 6.4M floats

    float* dense    = (float*)d_ws;                    // [0, 51.2MB)
    float* filtered = dense + denseElems;              // [51.2MB, 76.8MB)
    float* tmp      = dense;                           // reuse: dense dead after GEMM
    float* out      = (float*)d_out;

    // 1) zero dense feature matrix and the output accumulator
    zero4_kernel<<<2048, 256, 0, stream>>>((float4*)dense, denseElems / 4);
    zero4_kernel<<<2048, 256, 0, stream>>>((float4*)out, featElems / 4);

    // 2) densify sparse features (atomic scatter handles duplicate (r,c))
    scatter_features_kernel<<<(NNZ_CNT + 255) / 256, 256, 0, stream>>>(
        feat_idx, feat_idx + NNZ_CNT, feat_vals, dense, NNZ_CNT);

    // 3) filtered = dense @ W via WMMA (3125 strips, 4 waves/block -> 782 blocks)
    gemm_wmma_kernel<<<(3125 + 3) / 4, 128, 0, stream>>>(dense, W, filtered, N_NODES);

    // 4) zero tmp (overlaps dense region, which GEMM has finished reading)
    zero4_kernel<<<2048, 256, 0, stream>>>((float4*)tmp, featElems / 4);

    // 5) tmp = phi_inv @ filtered   (8 nnz per 256-thread block)
    spmm_scatter_kernel<<<(NNZ_CNT + 7) / 8, 256, 0, stream>>>(
        phinv_idx, phinv_idx + NNZ_CNT, phinv_vals, nullptr, filtered, tmp, NNZ_CNT);

    // 6) out = (phi * diag(theta)) @ tmp
    spmm_scatter_kernel<<<(NNZ_CNT + 7) / 8, 256, 0, stream>>>(
        phi_idx, phi_idx + NNZ_CNT, phi_vals, theta, tmp, out, NNZ_CNT);

    // 7) relu in place
    relu4_kernel<<<2048, 256, 0, stream>>>((float4*)out, featElems / 4);
}